// Model_1735166788454
// MI455X (gfx1250) — compile-verified
//
#include <hip/hip_runtime.h>

typedef __attribute__((ext_vector_type(2))) float v2f;
typedef __attribute__((ext_vector_type(8))) float v8f;
struct __attribute__((packed, aligned(4))) f2 { float x, y; };   // 8B, 4B-aligned load

// problem geometry
#define XN 8
#define XC 3
#define XD 40
#define XH 128
#define XW 128
#define OC 16
#define DO_ 9
#define HO 31
#define WO 31
#define WB 8            // blocks of 16 w-positions
#define KTOT 81
#define KSTEPS 21

// strides (dwords)
#define XSC (XD*XH*XW)      // 655360
#define XSN (XC*XSC)        // 1966080
#define XSD (XH*XW)         // 16384
#define XSH (XW)            // 128
#define OSC (DO_*HO*WO)     // 8649
#define OSN (OC*OSC)        // 138384
#define OSD (HO*WO)         // 961
#define OSH (WO)            // 31

#define WMMA4(A, B, C) __builtin_amdgcn_wmma_f32_16x16x4_f32(false, (A), false, (B), (short)0, (C), false, false)

// load one tile's A fragments (29 requests: 13x b64 + 16x b32), uniform base in SGPRs
#define LOADT(P64, PS0, PS1, XBB) do {                                          \
  _Pragma("unroll") for (int _s = 0; _s < 13; ++_s)                             \
    P64[_s] = *(const f2*)((XBB) + voff64[_s]);                                 \
  _Pragma("unroll") for (int _i = 0; _i < 8; ++_i) {                            \
    PS0[_i] = *(const float*)((XBB) + voffs0[_i]);                              \
    PS1[_i] = *(const float*)((XBB) + voffs1[_i]);                              \
  }                                                                             \
} while (0)

// 21 WMMAs + bias + channel softmax (16-lane halves) + w-pool, accumulate d/h pool
#define PROC(P64, PS0, PS1) do {                                                \
  v8f c = {};                                                                   \
  _Pragma("unroll") for (int _s = 0; _s < 13; ++_s) {                           \
    v2f a;  a[0] = P64[_s].x; a[1] = P64[_s].y;                                 \
    v2f bm; bm[0] = bw0[_s];  bm[1] = bw1[_s];                                  \
    c = WMMA4(a, bm, c);                                                        \
  }                                                                             \
  _Pragma("unroll") for (int _s = 13; _s < 21; ++_s) {                          \
    v2f a;  a[0] = PS0[_s-13]; a[1] = PS1[_s-13];                               \
    v2f bm; bm[0] = bw0[_s];   bm[1] = bw1[_s];                                 \
    c = WMMA4(a, bm, c);                                                        \
  }                                                                             \
  _Pragma("unroll") for (int _v = 0; _v < 8; ++_v) {                            \
    float y = c[_v] + bch;                                                      \
    float m = y;                                                                \
    _Pragma("unroll") for (int _d = 1; _d < 16; _d <<= 1)                       \
      m = fmaxf(m, __shfl_xor(m, _d, 16));                                      \
    float e = __expf(y - m);                                                    \
    float ss = e;                                                               \
    _Pragma("unroll") for (int _d = 1; _d < 16; _d <<= 1)                       \
      ss += __shfl_xor(ss, _d, 16);                                             \
    c[_v] = e * __builtin_amdgcn_rcpf(ss);                                      \
  }                                                                             \
  pA = fmaxf(pA, fmaxf(fmaxf(c[0], c[1]), fmaxf(c[2], c[3])));                  \
  pB = fmaxf(pB, fmaxf(fmaxf(c[4], c[5]), fmaxf(c[6], c[7])));                  \
} while (0)

__global__ __launch_bounds__(256) void conv_softmax_pool_kernel(
    const float* __restrict__ x, const float* __restrict__ Wt,
    const float* __restrict__ bias, float* __restrict__ out)
{
  const int lane = threadIdx.x & 31;
  const int wave = threadIdx.x >> 5;
  const int half = lane >> 4;      // selects K slots {4s,4s+1} vs {4s+2,4s+3}
  const int ln   = lane & 15;      // A: M row; B/C: channel

  // wave-uniform cell id -> scalar regs -> saddr-form loads
  const int cell = __builtin_amdgcn_readfirstlane((int)(blockIdx.x * 8 + wave));
  int wb = cell & 7;
  int t  = cell >> 3;
  int ho = t % HO; t /= HO;
  int dO = t % DO_;
  int n  = t / DO_;

  const int w = wb * 16 + ln;      // this lane's output-w row (M)

  // ---- k-slot assignment (A and B agree):
  //   s=0..12 : b64 pair = group G=2s+half, kw={0,1}   (contiguous in w)
  //   s=13..20: singles u=(s-13)*4+2*half+p over [(26,0),(26,1),(26,2),(g,2) g=0..25, pad x3]
  float bw0[KSTEPS], bw1[KSTEPS];
  int voff64[13], voffs0[8], voffs1[8];
#pragma unroll
  for (int s = 0; s < 13; ++s) {
    int G  = 2 * s + half;
    int ci = G / 9, kd = (G % 9) / 3, kh = G % 3;
    voff64[s] = (w + ci * XSC + kd * XSD + kh * XSH) * 4;
    int kc = ci * 27 + kd * 9 + kh * 3;
    bw0[s] = Wt[ln * KTOT + kc];
    bw1[s] = Wt[ln * KTOT + kc + 1];
  }
#pragma unroll
  for (int i = 0; i < 8; ++i) {
#pragma unroll
    for (int p = 0; p < 2; ++p) {
      int u = i * 4 + 2 * half + p;
      int g, kw, valid = 1;
      if      (u == 0)  { g = 26;    kw = 0; }
      else if (u == 1)  { g = 26;    kw = 1; }
      else if (u == 2)  { g = 26;    kw = 2; }
      else if (u <= 28) { g = u - 3; kw = 2; }
      else              { g = 0;     kw = 0; valid = 0; }
      int ci = g / 9, kd = (g % 9) / 3, kh = g % 3;
      int vo = (w + (valid ? (ci * XSC + kd * XSD + kh * XSH + kw) : 0)) * 4;
      float bv = valid ? Wt[ln * KTOT + ci * 27 + kd * 9 + kh * 3 + kw] : 0.0f;
      if (p == 0) { voffs0[i] = vo; bw0[13 + i] = bv; }
      else        { voffs1[i] = vo; bw1[13 + i] = bv; }
    }
  }
  const float bch = bias[ln];

  const char* xn = (const char*)(x + (size_t)n * XSN);
#define TB(T) (xn + (size_t)4 * (size_t)((dO * 4 + ((T) >> 2)) * XSD + (ho * 4 + ((T) & 3)) * XSH))

  float pA = 0.0f, pB = 0.0f;      // softmax outputs > 0

  // double-buffered A fragments, software-pipelined over the 16 (dd,hh) tiles
  f2 Aa64[13]; float Aas0[8], Aas1[8];
  f2 Ab64[13]; float Abs0[8], Abs1[8];

  LOADT(Aa64, Aas0, Aas1, TB(0));
  for (int tt = 0; tt < 16; tt += 2) {
    LOADT(Ab64, Abs0, Abs1, TB(tt + 1));     // prefetch tile tt+1
    PROC(Aa64, Aas0, Aas1);                  // consume tile tt
    if (tt + 2 < 16) {
      LOADT(Aa64, Aas0, Aas1, TB(tt + 2));   // prefetch tile tt+2
    }
    PROC(Ab64, Abs0, Abs1);                  // consume tile tt+1
  }
#undef TB

  int wo0 = wb * 4 + 2 * half;
  long obase = (long)n * OSN + (long)ln * OSC + (long)dO * OSD + (long)ho * OSH;
  if (wo0 < WO)     out[obase + wo0]     = pA;
  if (wo0 + 1 < WO) out[obase + wo0 + 1] = pB;   // drops wo=31 (truncated pool region)
}

extern "C" void kernel_launch(void* const* d_in, const int* in_sizes, int n_in,
                              void* d_out, int out_size, void* d_ws, size_t ws_size,
                              hipStream_t stream) {
  (void)in_sizes; (void)n_in; (void)d_ws; (void)ws_size; (void)out_size;
  const float* x  = (const float*)d_in[0];
  const float* Wt = (const float*)d_in[1];
  const float* b  = (const float*)d_in[2];
  float* out = (float*)d_out;

  const int ncells = XN * DO_ * HO * WB;   // 17856 = 2232 * 8 waves/block (exact)
  dim3 grid(ncells / 8);
  dim3 block(256);
  conv_softmax_pool_kernel<<<grid, block, 0, stream>>>(x, Wt, b, out);
}